// ContactsFittingLoss_78125455114325
// MI455X (gfx1250) — compile-verified
//
#include <hip/hip_runtime.h>
#include <math.h>

// ---------------------------------------------------------------------------
// ContactsFittingLoss for MI455X (gfx1250, wave32).
// Heavy part: two (B,V,O) pairwise-distance sweeps done with
// V_WMMA_F32_16X16X4_F32. A preprocessing kernel packs, per 16-obj tile, the
// exact WMMA A-operand lane image for BOTH sweeps (coords with |o|^2 folded
// into the K=4 slot), so the hot loop is: async-DMA tile into LDS
// (global_load_async_to_lds_b128, double buffered, s_wait_asynccnt), one
// ds_load_b128 per lane, two WMMAs, branch-skipped top-5/argmin selection.
// ---------------------------------------------------------------------------

typedef float v2f __attribute__((ext_vector_type(2)));
typedef float v8f __attribute__((ext_vector_type(8)));

#define LOG2PI_F 1.8378770664093453f
#define WTHRESH  0.01f
#define ATHRESH  1e-9f
#define FINF     __builtin_inff()
#define BIGF     3.0e37f

__device__ __forceinline__ void cswap(float &a, float &b) {
  float lo = fminf(a, b);
  b = fmaxf(a, b);
  a = lo;
}
// insert x into ascending 5-list keeping the 5 smallest
__device__ __forceinline__ void ins5(float &s0, float &s1, float &s2,
                                     float &s3, float &s4, float x) {
  s4 = fminf(s4, x);
  cswap(s3, s4); cswap(s2, s3); cswap(s1, s2); cswap(s0, s1);
}

// gfx1250 async global->LDS copy (ASYNCcnt tracked), inline asm for
// portability across ROCm / amdgpu-toolchain clang builtin differences.
__device__ __forceinline__ void async_b128_to_lds(unsigned lds_off,
                                                  const void* gaddr) {
  asm volatile("global_load_async_to_lds_b128 %0, %1, off"
               :: "v"(lds_off), "v"(gaddr)
               : "memory");
}
__device__ __forceinline__ void wait_async_le1() {
  asm volatile("s_wait_asynccnt 0x1" ::: "memory");
}
__device__ __forceinline__ void wait_async_le0() {
  asm volatile("s_wait_asynccnt 0x0" ::: "memory");
}

// -------------------- gid[v] = argmin_g ||init_verts - init_anchors|| ------
__global__ void k_gid(const float* __restrict__ iv, const float* __restrict__ ia,
                      int* __restrict__ gid, int V, int G) {
  int v = blockIdx.x * blockDim.x + threadIdx.x;
  if (v >= V) return;
  float x = iv[v * 3 + 0], y = iv[v * 3 + 1], z = iv[v * 3 + 2];
  float best = FINF; int bg = 0;
  for (int g = 0; g < G; ++g) {
    float dx = x - ia[g * 3 + 0], dy = y - ia[g * 3 + 1], dz = z - ia[g * 3 + 2];
    float d2 = dx * dx + dy * dy + dz * dz;
    if (d2 < best) { best = d2; bg = g; }   // ascending scan -> first occurrence
  }
  gid[v] = bg;
}

// -------------------- active[g] = any |cg| > 1e-9 over (b, 12) -------------
__global__ void k_active(const float* __restrict__ cg, int* __restrict__ active,
                         int B, int G) {
  int g = threadIdx.x;
  if (g >= G) return;
  int a = 0;
  for (int b = 0; b < B; ++b)
    for (int j = 0; j < 12; ++j)
      if (fabsf(cg[((size_t)b * G + g) * 12 + j]) > ATHRESH) a = 1;
  active[g] = a;
}

// -------------------- per (b,g): mean, Cholesky L, half-logdet -------------
__global__ void k_gauss(const float* __restrict__ cg,
                        const float* __restrict__ anchors,
                        float* __restrict__ gauss, int B, int G) {
  int i = blockIdx.x * blockDim.x + threadIdx.x;
  if (i >= B * G) return;
  const float* c = cg + (size_t)i * 12;
  const float* a = anchors + (size_t)i * 3;
  float mx = c[0] + a[0], my = c[1] + a[1], mz = c[2] + a[2];
  float a00 = c[3], a10 = c[6], a11 = c[7], a20 = c[9], a21 = c[10], a22 = c[11];
  float L00 = sqrtf(a00);
  float L10 = a10 / L00, L20 = a20 / L00;
  float L11 = sqrtf(a11 - L10 * L10);
  float L21 = (a21 - L20 * L10) / L11;
  float L22 = sqrtf(a22 - L20 * L20 - L21 * L21);
  float hld = logf(L00) + logf(L11) + logf(L22);
  float* o = gauss + (size_t)i * 10;
  o[0] = mx; o[1] = my; o[2] = mz;
  o[3] = L00; o[4] = L10; o[5] = L11; o[6] = L20; o[7] = L21; o[8] = L22;
  o[9] = hld;
}

// -------------------- w[b,v] = exp(logp of selected gaussian) --------------
__global__ void k_w(const float* __restrict__ verts, const float* __restrict__ gauss,
                    const int* __restrict__ gid, float* __restrict__ wbuf,
                    int B, int V, int G) {
  int i = blockIdx.x * blockDim.x + threadIdx.x;
  if (i >= B * V) return;
  int b = i / V, v = i % V;
  const float* gp = gauss + ((size_t)b * G + gid[v]) * 10;
  const float* vp = verts + (size_t)i * 3;
  float d0 = vp[0] - gp[0], d1 = vp[1] - gp[1], d2 = vp[2] - gp[2];
  float y0 = d0 / gp[3];
  float y1 = (d1 - gp[4] * y0) / gp[5];
  float y2 = (d2 - gp[6] * y0 - gp[7] * y1) / gp[8];
  float maha = y0 * y0 + y1 * y1 + y2 * y2;
  wbuf[i] = expf(-0.5f * (maha + 3.0f * LOG2PI_F) - gp[9]);
}

// -------------------- per (b,g) segment min/max of w -----------------------
__global__ __launch_bounds__(128) void k_seg(const float* __restrict__ wbuf,
                                             const int* __restrict__ gid,
                                             float* __restrict__ segmn,
                                             float* __restrict__ segmx,
                                             int B, int V, int G) {
  __shared__ float smn[128], smx[128];
  int bg = blockIdx.x;
  int b = bg / G, g = bg % G;
  float mn = FINF, mx = -FINF;
  for (int v = threadIdx.x; v < V; v += 128) {
    if (gid[v] == g) {
      float w = wbuf[(size_t)b * V + v];
      mn = fminf(mn, w); mx = fmaxf(mx, w);
    }
  }
  smn[threadIdx.x] = mn; smx[threadIdx.x] = mx;
  __syncthreads();
  for (int s = 64; s > 0; s >>= 1) {
    if (threadIdx.x < s) {
      smn[threadIdx.x] = fminf(smn[threadIdx.x], smn[threadIdx.x + s]);
      smx[threadIdx.x] = fmaxf(smx[threadIdx.x], smx[threadIdx.x + s]);
    }
    __syncthreads();
  }
  if (threadIdx.x == 0) { segmn[bg] = smn[0]; segmx[bg] = smx[0]; }
}

// -------------------- pack WMMA A-operand tiles ----------------------------
// pack[b][t][lane][4]:
//   lane l  (<16): (P.x, P.y, N.x, N.y)           (K0,K1 halves of both sweeps)
//   lane l+16    : (P.z, |P|^2, N.z, |N|^2)       (K2,K3 halves; norm in K3)
// Padded rows (o >= O) get coords 0 / norm BIG -> huge d^2, never selected.
__global__ void k_pack(const float* __restrict__ obj_pts,
                       const float* __restrict__ obj_nrm,
                       float* __restrict__ pack,
                       int B, int O, int NTO) {
  int i = blockIdx.x * blockDim.x + threadIdx.x;   // over B*NTO*16
  if (i >= B * NTO * 16) return;
  int col = i & 15;
  int t = (i >> 4) % NTO;
  int b = (i >> 4) / NTO;
  int o = t * 16 + col;
  float px = 0.f, py = 0.f, pz = 0.f, pn = BIGF;
  float nx = 0.f, ny = 0.f, nz = 0.f, nn = BIGF;
  if (o < O) {
    const float* pp = obj_pts + ((size_t)b * O + o) * 3;
    px = pp[0]; py = pp[1]; pz = pp[2];
    pn = px * px + py * py + pz * pz;
    const float* np = obj_nrm + ((size_t)b * O + o) * 6;
    nx = np[0]; ny = np[1]; nz = np[2];
    nn = nx * nx + ny * ny + nz * nz;
  }
  float* base = pack + (size_t)(b * NTO + t) * 128;   // 32 lanes * 4 floats
  *(float4*)(base + (size_t)col * 4)        = make_float4(px, py, nx, ny);
  *(float4*)(base + (size_t)(col + 16) * 4) = make_float4(pz, pn, nz, nn);
}

// -------------------- the WMMA distance sweep ------------------------------
// block = 256 threads = 8 waves; one block per (b, 16-vert tile).
// Each wave handles a contiguous ascending range of 16-obj tiles; packed
// A-tiles are async-DMA'd into a per-wave LDS double buffer.
__global__ __launch_bounds__(256) void k_main(
    const float* __restrict__ verts,
    const float* __restrict__ pack,
    float* __restrict__ knnsum,
    int* __restrict__ nnidx,
    int B, int V, int O, int NTV, int NTO)
{
  __shared__ __align__(16) float shA[8][2][128];  // [wave][parity][lane*4]
  __shared__ float shT[8][16][5];
  __shared__ float shD[8][16];
  __shared__ int   shI[8][16];

  const int lane = threadIdx.x & 31;
  const int wv   = threadIdx.x >> 5;   // wave 0..7
  const int col  = lane & 15;          // vert column within tile
  const int kh   = lane >> 4;          // 0: K0/K1 half, 1: K2/K3 half
  const int b    = blockIdx.x / NTV;
  const int vt   = blockIdx.x % NTV;
  const int v    = vt * 16 + col;

  // B operand (verts), fixed for the whole block.
  float vx = 0.f, vy = 0.f, vz = 0.f;
  if (v < V) {
    const float* vp = verts + ((size_t)b * V + v) * 3;
    vx = vp[0]; vy = vp[1]; vz = vp[2];
  }
  float vpart = kh ? (vz * vz) : (vx * vx + vy * vy);
  float vnorm = vpart + __shfl_xor(vpart, 16, 32);
  v2f Bv;
  Bv.x = kh ? (-2.f * vz) : (-2.f * vx);
  Bv.y = kh ? 1.0f        : (-2.f * vy);   // K3 column of B is all ones

  const int TPW  = (NTO + 7) >> 3;
  const int tBeg = wv * TPW;
  int tEnd = tBeg + TPW; if (tEnd > NTO) tEnd = NTO;

  float s0 = FINF, s1 = FINF, s2 = FINF, s3 = FINF, s4 = FINF; // 5 smallest d^2
  float bestd = FINF; int besti = 0;                            // argmin sweep
  const v8f cz = {0.f, 0.f, 0.f, 0.f, 0.f, 0.f, 0.f, 0.f};

  const unsigned ldsLane0 = (unsigned)(uintptr_t)&shA[wv][0][0] + (unsigned)lane * 16u;
  const unsigned ldsLane1 = (unsigned)(uintptr_t)&shA[wv][1][0] + (unsigned)lane * 16u;
  const float* gLane = pack + (size_t)(b * NTO) * 128 + (size_t)lane * 4;

  if (tBeg < tEnd) {
    // preload first tile into parity 0
    async_b128_to_lds(ldsLane0, gLane + (size_t)tBeg * 128);

    for (int t = tBeg; t < tEnd; ++t) {
      const int par = (t - tBeg) & 1;
      if (t + 1 < tEnd) {
        async_b128_to_lds(par ? ldsLane0 : ldsLane1, gLane + (size_t)(t + 1) * 128);
        wait_async_le1();   // tile t resident, t+1 in flight
      } else {
        wait_async_le0();
      }
      const float4 q = *(const float4*)&shA[wv][par][lane * 4];
      v2f Ap; Ap.x = q.x; Ap.y = q.y;   // obj_pts sweep
      v2f An; An.x = q.z; An.y = q.w;   // obj_normals sweep

      // ---- sweep 1: top-5 squared distances to obj_pts ----
      {
        v8f c = __builtin_amdgcn_wmma_f32_16x16x4_f32(false, Ap, false, Bv,
                                                      (short)0, cz, false, false);
        float d0 = c[0] + vnorm, d1 = c[1] + vnorm, d2 = c[2] + vnorm, d3 = c[3] + vnorm;
        float d4 = c[4] + vnorm, d5 = c[5] + vnorm, d6 = c[6] + vnorm, d7 = c[7] + vnorm;
        float m = fminf(fminf(fminf(d0, d1), fminf(d2, d3)),
                        fminf(fminf(d4, d5), fminf(d6, d7)));
        if (m < s4) {   // rare after warm-up: tile contributes to top-5
          ins5(s0, s1, s2, s3, s4, d0); ins5(s0, s1, s2, s3, s4, d1);
          ins5(s0, s1, s2, s3, s4, d2); ins5(s0, s1, s2, s3, s4, d3);
          ins5(s0, s1, s2, s3, s4, d4); ins5(s0, s1, s2, s3, s4, d5);
          ins5(s0, s1, s2, s3, s4, d6); ins5(s0, s1, s2, s3, s4, d7);
        }
      }

      // ---- sweep 2: argmin squared distance to obj_normals[:, :3] ----
      {
        v8f c = __builtin_amdgcn_wmma_f32_16x16x4_f32(false, An, false, Bv,
                                                      (short)0, cz, false, false);
        float e0 = c[0] + vnorm, e1 = c[1] + vnorm, e2 = c[2] + vnorm, e3 = c[3] + vnorm;
        float e4 = c[4] + vnorm, e5 = c[5] + vnorm, e6 = c[6] + vnorm, e7 = c[7] + vnorm;
        float m = fminf(fminf(fminf(e0, e1), fminf(e2, e3)),
                        fminf(fminf(e4, e5), fminf(e6, e7)));
        if (m < bestd) {   // strict < keeps earliest obj index on ties
          bestd = m;
          int rr = 7;      // smallest r attaining m (scan high->low)
          rr = (e6 == m) ? 6 : rr;
          rr = (e5 == m) ? 5 : rr;
          rr = (e4 == m) ? 4 : rr;
          rr = (e3 == m) ? 3 : rr;
          rr = (e2 == m) ? 2 : rr;
          rr = (e1 == m) ? 1 : rr;
          rr = (e0 == m) ? 0 : rr;
          besti = t * 16 + rr + 8 * kh;
        }
      }
    }
  }

  // merge the two row-halves (lane <-> lane^16, same vert column)
  {
    float o0 = __shfl_xor(s0, 16, 32);
    float o1 = __shfl_xor(s1, 16, 32);
    float o2 = __shfl_xor(s2, 16, 32);
    float o3 = __shfl_xor(s3, 16, 32);
    float o4 = __shfl_xor(s4, 16, 32);
    ins5(s0, s1, s2, s3, s4, o0); ins5(s0, s1, s2, s3, s4, o1);
    ins5(s0, s1, s2, s3, s4, o2); ins5(s0, s1, s2, s3, s4, o3);
    ins5(s0, s1, s2, s3, s4, o4);
    float od = __shfl_xor(bestd, 16, 32);
    int   oi = __shfl_xor(besti, 16, 32);
    bool better = (od < bestd) || (od == bestd && oi < besti);
    bestd = better ? od : bestd;
    besti = better ? oi : besti;
  }

  if (kh == 0) {
    shT[wv][col][0] = s0; shT[wv][col][1] = s1; shT[wv][col][2] = s2;
    shT[wv][col][3] = s3; shT[wv][col][4] = s4;
    shD[wv][col] = bestd; shI[wv][col] = besti;
  }
  __syncthreads();

  // merge the 8 waves (ascending obj ranges -> tie-break keeps lowest index)
  if (threadIdx.x < 16) {
    const int c = threadIdx.x;
    float t0 = FINF, t1 = FINF, t2 = FINF, t3 = FINF, t4 = FINF;
    float bd = FINF; int bi = 0;
    for (int w = 0; w < 8; ++w) {
#pragma unroll
      for (int i = 0; i < 5; ++i) ins5(t0, t1, t2, t3, t4, shT[w][c][i]);
      float d = shD[w][c]; int ii = shI[w][c];
      bool better = (d < bd) || (d == bd && ii < bi);
      bd = better ? d : bd;
      bi = better ? ii : bi;
    }
    const int vv = vt * 16 + c;
    if (vv < V) {
      float s = sqrtf(fmaxf(t0, 0.f)) + sqrtf(fmaxf(t1, 0.f)) +
                sqrtf(fmaxf(t2, 0.f)) + sqrtf(fmaxf(t3, 0.f)) +
                sqrtf(fmaxf(t4, 0.f));
      knnsum[(size_t)b * V + vv] = s;
      nnidx[(size_t)b * V + vv] = bi;
    }
  }
}

// -------------------- final deterministic reduction ------------------------
__global__ __launch_bounds__(256) void k_final(
    const float* __restrict__ verts,
    const float* __restrict__ obj_nrm,
    const float* __restrict__ wbuf,
    const float* __restrict__ knnsum,
    const int* __restrict__ nnidx,
    const int* __restrict__ gid,
    const int* __restrict__ active,
    const float* __restrict__ segmn,
    const float* __restrict__ segmx,
    float* __restrict__ out,
    int B, int V, int O, int G)
{
  __shared__ float r1[256], r2[256];
  float acc1 = 0.f, acc2 = 0.f;
  const int N = B * V;
  for (int i = threadIdx.x; i < N; i += 256) {
    const int b = i / V;
    const int g = gid[i % V];
    float w  = wbuf[i];
    float mn = segmn[b * G + g];
    float mx = segmx[b * G + g];
    float wn = (w - mn) / (mx - mn);     // NaN when mx==mn -> compare false
    wn = (wn > WTHRESH) ? wn : 0.f;
    float eff = active[g] ? wn : 0.f;
    acc1 += eff * knnsum[i];

    const float* vp = verts + (size_t)i * 3;
    const float* np = obj_nrm + ((size_t)b * O + nnidx[i]) * 6;
    float dx = vp[0] - np[0], dy = vp[1] - np[1], dz = vp[2] - np[2];
    float nx = np[3], ny = np[4], nz = np[5];
    // dp = n . (v - (p - 0.002 n)) = n.(v-p) + 0.002 |n|^2
    float dp = nx * dx + ny * dy + nz * dz + 0.002f * (nx * nx + ny * ny + nz * nz);
    acc2 += fmaxf(0.f, -dp);
  }
  r1[threadIdx.x] = acc1; r2[threadIdx.x] = acc2;
  __syncthreads();
  for (int s = 128; s > 0; s >>= 1) {
    if (threadIdx.x < s) {
      r1[threadIdx.x] += r1[threadIdx.x + s];
      r2[threadIdx.x] += r2[threadIdx.x + s];
    }
    __syncthreads();
  }
  if (threadIdx.x == 0) {
    out[0] = r1[0] / (float)(N * 5);   // K = 5
    out[1] = r2[0] / (float)N;
  }
}

// ---------------------------------------------------------------------------
extern "C" void kernel_launch(void* const* d_in, const int* in_sizes, int n_in,
                              void* d_out, int out_size, void* d_ws, size_t ws_size,
                              hipStream_t stream)
{
  const float* verts        = (const float*)d_in[0];
  const float* anchor_verts = (const float*)d_in[1];
  const float* obj_pts      = (const float*)d_in[2];
  const float* cg           = (const float*)d_in[3];
  const float* obj_nrm      = (const float*)d_in[4];
  const float* init_verts   = (const float*)d_in[5];
  const float* init_anchors = (const float*)d_in[6];
  (void)n_in; (void)out_size; (void)ws_size;

  const int V = in_sizes[5] / 3;                 // 778
  const int G = in_sizes[6] / 3;                 // 32
  const int B = in_sizes[0] / (3 * V);           // 16
  const int O = in_sizes[2] / (3 * B);           // 8192
  const int NTV = (V + 15) / 16;
  const int NTO = (O + 15) / 16;

  // workspace carve-out (256B aligned slabs)
  char* p = (char*)d_ws;
  auto alloc = [&](size_t bytes) -> void* {
    void* r = (void*)p;
    p += (bytes + 255) & ~(size_t)255;
    return r;
  };
  int*   gid    = (int*)  alloc(sizeof(int)   * (size_t)V);
  int*   nnidx  = (int*)  alloc(sizeof(int)   * (size_t)B * V);
  int*   active = (int*)  alloc(sizeof(int)   * (size_t)G);
  float* gauss  = (float*)alloc(sizeof(float) * (size_t)B * G * 10);
  float* wbuf   = (float*)alloc(sizeof(float) * (size_t)B * V);
  float* segmn  = (float*)alloc(sizeof(float) * (size_t)B * G);
  float* segmx  = (float*)alloc(sizeof(float) * (size_t)B * G);
  float* knns   = (float*)alloc(sizeof(float) * (size_t)B * V);
  float* pack   = (float*)alloc(sizeof(float) * (size_t)B * NTO * 128);

  k_gid   <<<(V + 255) / 256, 256, 0, stream>>>(init_verts, init_anchors, gid, V, G);
  k_active<<<1, 64, 0, stream>>>(cg, active, B, G);
  k_gauss <<<(B * G + 127) / 128, 128, 0, stream>>>(cg, anchor_verts, gauss, B, G);
  k_w     <<<(B * V + 255) / 256, 256, 0, stream>>>(verts, gauss, gid, wbuf, B, V, G);
  k_seg   <<<B * G, 128, 0, stream>>>(wbuf, gid, segmn, segmx, B, V, G);
  k_pack  <<<(B * NTO * 16 + 255) / 256, 256, 0, stream>>>(obj_pts, obj_nrm, pack, B, O, NTO);

  k_main  <<<B * NTV, 256, 0, stream>>>(verts, pack, knns, nnidx, B, V, O, NTV, NTO);
  k_final <<<1, 256, 0, stream>>>(verts, obj_nrm, wbuf, knns, nnidx, gid, active,
                                  segmn, segmx, (float*)d_out, B, V, O, G);
}